// VonNeumannEntropy_82480551953010
// MI455X (gfx1250) — compile-verified
//
#include <hip/hip_runtime.h>
#include <hip/hip_bf16.h>

// VonNeumann entanglement entropy, MI455X (gfx1250, wave32).
// rho_A has rank <= d_B (2/3/4), so eigvalsh(128/168/256) collapses to the
// tiny Gram matrix G = psi_m^T psi_m. G is computed with V_WMMA_F32_16X16X4_F32
// (A register image reused as B == A^T per ISA 7.12.2 layouts => D = A A^T),
// 4 batch elements per 16x16 tile. Eigen + entropy done in f32 VALU.

typedef __attribute__((ext_vector_type(2))) float v2f;
typedef __attribute__((ext_vector_type(8))) float v8f;

#define EPS_TRACE 1e-10f
#define EPS_EIG   1e-10f
// -EPS_EIG * ln(EPS_EIG): contribution of each zero (clamped) eigenvalue
#define PAD_TERM  2.30258509e-9f

// D-tile element (m,n) from the 8x32 LDS image of the WMMA C/D layout:
// lanes 0-15 hold rows M=0..7 (vgpr r = row r), lanes 16-31 rows M=8..15.
__device__ __forceinline__ float D_at(const float* tile, int m, int n) {
  return tile[(m & 7) * 32 + ((m & 8) << 1) + n];
}

// Accumulate G-tile: A[m,kk] = psi_b(m)[(4s+kk)*DB + j(m)], m = DB*bl + j.
// A 16x4 f32 layout: lane<16 holds (M=lane, K=0/1) in v0/v1; lane>=16 K=2/3.
template<int DB, int STEPS>
__device__ __forceinline__ void gram_wmma(const float* row, int half, int j,
                                          float* tile, int lane) {
  v8f acc = {};
  for (int s = 0; s < STEPS; ++s) {
    const int base = (4 * s + 2 * half) * DB + j;
    v2f a;
    a[0] = row[base];
    a[1] = row[base + DB];
    // D = A * A^T (same regs as A and B); 8-arg f32 form:
    // (neg_a, A, neg_b, B, c_mod, C, reuse_a, reuse_b)
    acc = __builtin_amdgcn_wmma_f32_16x16x4_f32(
        false, a, false, a, (short)0, acc, false, false);
  }
#pragma unroll
  for (int r = 0; r < 8; ++r) tile[r * 32 + lane] = acc[r];
}

// Eigenvalues of 4x4 symmetric via cyclic Jacobi (6 sweeps, values only).
__device__ __forceinline__ void jacobi4(float A[4][4], float ev[4]) {
#pragma unroll
  for (int sweep = 0; sweep < 6; ++sweep) {
#pragma unroll
    for (int pi = 0; pi < 6; ++pi) {
      const int Pp[6] = {0, 0, 0, 1, 1, 2};
      const int Qq[6] = {1, 2, 3, 2, 3, 3};
      const int p = Pp[pi], q = Qq[pi];
      float apq = A[p][q];
      if (fabsf(apq) > 1e-36f) {
        float app = A[p][p], aqq = A[q][q];
        float theta = (aqq - app) / (2.0f * apq);
        float t = 1.0f / (fabsf(theta) + sqrtf(theta * theta + 1.0f));
        t = (theta >= 0.0f) ? t : -t;
        float c = 1.0f / sqrtf(t * t + 1.0f);
        float s = t * c;
        A[p][p] = app - t * apq;
        A[q][q] = aqq + t * apq;
        A[p][q] = 0.0f;
        A[q][p] = 0.0f;
#pragma unroll
        for (int i = 0; i < 4; ++i) {
          if (i == p || i == q) continue;
          float aip = A[i][p], aiq = A[i][q];
          float nip = c * aip - s * aiq;
          float niq = c * aiq + s * aip;
          A[i][p] = nip; A[p][i] = nip;
          A[i][q] = niq; A[q][i] = niq;
        }
      }
    }
  }
  ev[0] = A[0][0]; ev[1] = A[1][1]; ev[2] = A[2][2]; ev[3] = A[3][3];
}

// Eigenvalues of 3x3 symmetric (trigonometric / Cardano).
__device__ __forceinline__ void eig3(float a00, float a01, float a02,
                                     float a11, float a12, float a22,
                                     float ev[3]) {
  float p1 = a01 * a01 + a02 * a02 + a12 * a12;
  float q = (a00 + a11 + a22) * (1.0f / 3.0f);
  float b00 = a00 - q, b11 = a11 - q, b22 = a22 - q;
  float p2 = b00 * b00 + b11 * b11 + b22 * b22 + 2.0f * p1;
  if (p2 < 1e-30f) { ev[0] = a00; ev[1] = a11; ev[2] = a22; return; }
  float p = sqrtf(p2 * (1.0f / 6.0f));
  float ip = 1.0f / p;
  float c00 = b00 * ip, c11 = b11 * ip, c22 = b22 * ip;
  float c01 = a01 * ip, c02 = a02 * ip, c12 = a12 * ip;
  float detB = c00 * (c11 * c22 - c12 * c12)
             - c01 * (c01 * c22 - c12 * c02)
             + c02 * (c01 * c12 - c11 * c02);
  float r = fminf(fmaxf(0.5f * detB, -1.0f), 1.0f);
  float phi = acosf(r) * (1.0f / 3.0f);
  float e0 = q + 2.0f * p * cosf(phi);
  float e2 = q + 2.0f * p * cosf(phi + 2.09439510f);  // + 2*pi/3
  float e1 = 3.0f * q - e0 - e2;
  ev[0] = e0; ev[1] = e1; ev[2] = e2;
}

// Entropy of one partition from raw-Gram eigenvalues.
// lam_i = (ev_i/nrm2) / (tr/nrm2 + EPS_TRACE), clamped to EPS_EIG;
// padn zero-eigenvalues contribute PAD_TERM each.
template<int N>
__device__ __forceinline__ float ent_part(const float* ev, float tr, float inv,
                                          float padn) {
  float denom = tr * inv + EPS_TRACE;
  float scale = inv / denom;
  float H = 0.0f;
#pragma unroll
  for (int i = 0; i < N; ++i) {
    float lam = fmaxf(ev[i] * scale, EPS_EIG);
    H -= lam * logf(lam);
  }
  return H + padn * PAD_TERM;
}

__global__ __launch_bounds__(128)
void vn_entropy_kernel(const float* __restrict__ sites, float* __restrict__ out) {
  __shared__ float psiLDS[16][512];          // 16 psi rows per block, 32 KB
  __shared__ float tiles[4][3][8][32];       // per-wave D tiles, 12 KB

  const int tid = threadIdx.x;
  const int lane = tid & 31;
  const int wv = tid >> 5;                   // wave 0..3, handles 4 batch rows
  const int bbase = blockIdx.x * 16;

  // Stage psi rows: psi[b][n*64+d] = sites[n][b][d]  (moveaxis gather),
  // coalesced float4 loads, 2048 float4 per block.
#pragma unroll
  for (int it = 0; it < 16; ++it) {
    int q = tid + it * 128;
    int r = q >> 7;            // local row 0..15
    int c4 = q & 127;          // float4 column
    int n = c4 >> 4;           // site chunk 0..7
    int d = (c4 & 15) * 4;     // element within chunk
    const float4 v = *reinterpret_cast<const float4*>(
        sites + (size_t)n * (512 * 64) + (size_t)(bbase + r) * 64 + d);
    *reinterpret_cast<float4*>(&psiLDS[r][c4 * 4]) = v;
  }
  __syncthreads();

  const int half = lane >> 4;
  const int lr = lane & 15;

  {  // partition f=0.25: d_B=4, 128 contraction rows -> 32 K-steps
    int bl = lr >> 2, j = lr & 3;
    gram_wmma<4, 32>(&psiLDS[wv * 4 + bl][0], half, j, &tiles[wv][0][0][0], lane);
  }
  {  // partition f=0.33: d_B=3, 168 rows (504 elems) -> 42 K-steps
    int bl = lr / 3, j = lr - bl * 3;
    if (bl > 3) { bl = 3; j = 2; }           // rows 12-15 unused, keep finite
    gram_wmma<3, 42>(&psiLDS[wv * 4 + bl][0], half, j, &tiles[wv][1][0][0], lane);
  }
  {  // partition f=0.5: d_B=2, 256 rows -> 64 K-steps
    int bl = lr >> 1, j = lr & 1;
    if (bl > 3) { bl = 3; j = 1; }           // rows 8-15 unused
    gram_wmma<2, 64>(&psiLDS[wv * 4 + bl][0], half, j, &tiles[wv][2][0][0], lane);
  }
  __syncthreads();

  if (lane < 4) {                            // one batch element per lane
    const int bl = lane;
    const float* tA = &tiles[wv][0][0][0];
    const float* tB = &tiles[wv][1][0][0];
    const float* tC = &tiles[wv][2][0][0];

    // d_B=4 Gram; its trace over all 512 components == ||psi_raw||^2
    float G[4][4];
#pragma unroll
    for (int jj = 0; jj < 4; ++jj)
#pragma unroll
      for (int kk = 0; kk < 4; ++kk)
        G[jj][kk] = D_at(tA, 4 * bl + jj, 4 * bl + kk);
    float S = G[0][0] + G[1][1] + G[2][2] + G[3][3];
    float inv = 1.0f / fmaxf(S, 1e-24f);     // max(norm, EPS_NORM)^2

    float ev4[4];
    jacobi4(G, ev4);
    float H = ent_part<4>(ev4, S, inv, 124.0f);       // 128-4 padded eigs

    float b00 = D_at(tB, 3 * bl + 0, 3 * bl + 0);
    float b01 = D_at(tB, 3 * bl + 0, 3 * bl + 1);
    float b02 = D_at(tB, 3 * bl + 0, 3 * bl + 2);
    float b11 = D_at(tB, 3 * bl + 1, 3 * bl + 1);
    float b12 = D_at(tB, 3 * bl + 1, 3 * bl + 2);
    float b22 = D_at(tB, 3 * bl + 2, 3 * bl + 2);
    float ev3[3];
    eig3(b00, b01, b02, b11, b12, b22, ev3);
    H += ent_part<3>(ev3, b00 + b11 + b22, inv, 165.0f);  // 168-3

    float c00 = D_at(tC, 2 * bl + 0, 2 * bl + 0);
    float c01 = D_at(tC, 2 * bl + 0, 2 * bl + 1);
    float c11 = D_at(tC, 2 * bl + 1, 2 * bl + 1);
    float mean = 0.5f * (c00 + c11);
    float disc = sqrtf(fmaxf(0.25f * (c00 - c11) * (c00 - c11) + c01 * c01, 0.0f));
    float ev2[2] = {mean + disc, mean - disc};
    H += ent_part<2>(ev2, c00 + c11, inv, 254.0f);        // 256-2

    out[bbase + wv * 4 + bl] = H * (1.0f / 3.0f);
  }
}

extern "C" void kernel_launch(void* const* d_in, const int* in_sizes, int n_in,
                              void* d_out, int out_size, void* d_ws, size_t ws_size,
                              hipStream_t stream) {
  (void)in_sizes; (void)n_in; (void)out_size; (void)d_ws; (void)ws_size;
  const float* sites = (const float*)d_in[0];   // (8, 512, 64) f32
  float* out = (float*)d_out;                   // (512,) f32
  vn_entropy_kernel<<<dim3(32), dim3(128), 0, stream>>>(sites, out);
}